// GAT_39410619908367
// MI455X (gfx1250) — compile-verified
//
#include <hip/hip_runtime.h>
#include <math.h>

typedef __attribute__((ext_vector_type(2))) float v2f;
typedef __attribute__((ext_vector_type(4))) float v4f;
typedef __attribute__((ext_vector_type(8))) float v8f;
typedef __attribute__((ext_vector_type(4))) unsigned int u32x4;
typedef __attribute__((ext_vector_type(8))) int i32x8;
typedef __attribute__((ext_vector_type(4))) int i32x4;

#define NFEAT  128
#define HID1   64      // HEADS1*HID
#define H1H    8
#define C1C    8
#define OUTC   32
#define SLOPE  0.2f

#define USE_TDM __has_builtin(__builtin_amdgcn_tensor_load_to_lds)

#if USE_TDM
// ---------------------------------------------------------------------------
// TDM: DMA a contiguous run of `elems` 4-byte elements from global to LDS.
// D# per CDNA5 ISA: count=1, type=2 (image),
// data_size=4B, 2-D tile [elems x 1], groups 2/3 zero (<=2D tensor).
// Tracked by TENSORcnt; caller waits with s_wait_tensorcnt.
// This toolchain exposes the 6-arg builtin:
//   (uint32x4 g0, int32x8 g1, int32x4 g2, int32x4 g3, int32x8 g4, i32 cpol)
// ---------------------------------------------------------------------------
__device__ __forceinline__ void tdm_load_1d(const void* gsrc, void* lds_dst,
                                            unsigned elems) {
    unsigned long long ga = (unsigned long long)(uintptr_t)gsrc;
    unsigned lds = (unsigned)(unsigned long long)(uintptr_t)lds_dst; // LDS offset
    u32x4 g0;
    g0[0] = 1u;                                   // count=1 (user descriptor)
    g0[1] = lds;                                  // lds_addr (bytes)
    g0[2] = (unsigned)ga;                         // global_addr[31:0]
    g0[3] = (unsigned)(ga >> 32) | (2u << 30);    // global_addr[56:32] | type=2
    i32x8 g1;
    unsigned L = elems;                           // tile_dim0 (fits 16 bits)
    g1[0] = 0x00020000;                           // data_size=2 -> 4 bytes
    g1[1] = (int)((L & 0xFFFFu) << 16);           // tensor_dim0[15:0]  @63:48
    g1[2] = (int)(((L >> 16) & 0xFFFFu) | (1u << 16)); // dim0 hi | tensor_dim1=1
    g1[3] = (int)((L & 0xFFFFu) << 16);           // tile_dim0 = L      @127:112
    g1[4] = 1;                                    // tile_dim1=1, tile_dim2=0
    g1[5] = (int)L;                               // tensor_dim0_stride lo
    g1[6] = 0;
    g1[7] = 0;
    i32x4 z4 = {0, 0, 0, 0};
    i32x8 z8 = {0, 0, 0, 0, 0, 0, 0, 0};
    __builtin_amdgcn_tensor_load_to_lds(g0, g1, z4, z4, z8, 0);
}
#endif

// ---------------------------------------------------------------------------
// One-shot K-pair packing of W: Wp[k/2][n] = {W[k][n], W[k+1][n]} so the
// GEMM's B operand is a single aligned ds_load_b64.
// ---------------------------------------------------------------------------
template <int K, int N>
__global__ void pack_w(const float* __restrict__ W, v2f* __restrict__ Wp) {
    int t = blockIdx.x * blockDim.x + threadIdx.x;
    if (t >= (K / 2) * N) return;
    int k2 = t / N, n = t % N;
    v2f p;
    p.x = W[(2 * k2 + 0) * N + n];
    p.y = W[(2 * k2 + 1) * N + n];
    Wp[t] = p;
}

// ---------------------------------------------------------------------------
// Dense GEMM  O[M,N] = X[M,K] @ W[K,N]  with V_WMMA_F32_16X16X4_F32.
// One wave per 16-row tile. A tile + packed W staged in LDS via TDM.
// ---------------------------------------------------------------------------
template <int K, int N>
__global__ __launch_bounds__(32) void gemm16_wmma(const float* __restrict__ X,
                                                  const v2f* __restrict__ Wp,
                                                  float* __restrict__ O,
                                                  int M) {
    __shared__ __align__(16) float As[16 * K];
    __shared__ __align__(16) v2f  Bs[(K / 2) * N];
    const int lane = threadIdx.x;
    const int m0   = blockIdx.x * 16;
    if (m0 >= M) return;

#if USE_TDM
    tdm_load_1d(&X[(size_t)m0 * K], &As[0], 16 * K);   // contiguous 16 rows
    tdm_load_1d(&Wp[0], &Bs[0], K * N);                // packed weights
    __builtin_amdgcn_s_wait_tensorcnt(0);
#else
    for (int i = lane * 4; i < 16 * K; i += 32 * 4)
        *(v4f*)&As[i] = *(const v4f*)&X[(size_t)m0 * K + i];
    for (int i = lane * 2; i < (K / 2) * N; i += 32 * 2)
        *(v4f*)&Bs[i] = *(const v4f*)&Wp[i];
#endif
    __syncthreads();

    const int col = lane & 15;   // N-index for B/C/D; M-index for A
    const int kh  = lane >> 4;   // which K-half this lane holds

    constexpr int NT = N / 16;
    v8f acc[NT] = {};            // 16x16 f32 C/D tiles, 8 VGPRs each

    for (int k0 = 0; k0 < K; k0 += 4) {
        // A 16x4: lane holds row=col, {K = k0+2*kh, k0+2*kh+1} (ds_load_b64)
        v2f a = *(const v2f*)&As[col * K + k0 + 2 * kh];
#pragma unroll
        for (int t = 0; t < NT; ++t) {
            // B 4x16: packed pair for (k0+2*kh, +1) at column t*16+col
            v2f b = Bs[(k0 / 2 + kh) * N + t * 16 + col];
            acc[t] = __builtin_amdgcn_wmma_f32_16x16x4_f32(
                false, a, false, b, (short)0, acc[t], false, false);
        }
    }

    // D layout: vgpr i -> row (i + 8*kh), lane&15 -> column
#pragma unroll
    for (int t = 0; t < NT; ++t)
#pragma unroll
        for (int i = 0; i < 8; ++i)
            O[(size_t)(m0 + i + 8 * kh) * N + t * 16 + col] = acc[t][i];
}

// ---------------------------------------------------------------------------
// Layer-1 per-node scores + accumulator init. Thread per (node, head).
// ---------------------------------------------------------------------------
__global__ void escore_init1(const float* __restrict__ Hm,
                             const float* __restrict__ asrc,
                             const float* __restrict__ adst,
                             float* __restrict__ es, float* __restrict__ ed,
                             float* __restrict__ mx, float* __restrict__ den,
                             float* __restrict__ out, int Nn) {
    int t = blockIdx.x * blockDim.x + threadIdx.x;
    if (t >= Nn * H1H) return;
    int n = t / H1H, h = t % H1H;
    const float* hp = Hm + (size_t)n * HID1 + h * C1C;
    float s = 0.f, d = 0.f;
#pragma unroll
    for (int c = 0; c < C1C; ++c) {
        float v = hp[c];
        s += v * asrc[h * C1C + c];
        d += v * adst[h * C1C + c];
    }
    es[t] = s;
    ed[t] = d;
    mx[t] = -3.0e38f;
    den[t] = 0.f;
    float* op = out + (size_t)n * HID1 + h * C1C;
#pragma unroll
    for (int c = 0; c < C1C; ++c) op[c] = 0.f;
}

// Layer-2 variant (H=1, C=32). Thread per node; block 0 also zeroes pooled[32].
__global__ void escore_init2(const float* __restrict__ Hm,
                             const float* __restrict__ asrc,
                             const float* __restrict__ adst,
                             float* __restrict__ es, float* __restrict__ ed,
                             float* __restrict__ mx, float* __restrict__ den,
                             float* __restrict__ out, float* __restrict__ pooled,
                             int Nn) {
    int t = blockIdx.x * blockDim.x + threadIdx.x;
    if (blockIdx.x == 0 && threadIdx.x < OUTC) pooled[threadIdx.x] = 0.f;
    if (t >= Nn) return;
    const float* hp = Hm + (size_t)t * OUTC;
    float s = 0.f, d = 0.f;
#pragma unroll
    for (int c = 0; c < OUTC; ++c) {
        float v = hp[c];
        s += v * asrc[c];
        d += v * adst[c];
    }
    es[t] = s;
    ed[t] = d;
    mx[t] = -3.0e38f;
    den[t] = 0.f;
    float* op = out + (size_t)t * OUTC;
#pragma unroll
    for (int c = 0; c < OUTC; ++c) op[c] = 0.f;
}

// ---------------------------------------------------------------------------
// Edge pass A: segment-max of leaky-relu scores (HW f32 atomic max).
// Thread per edge, loop over heads. Self-loops appended: e >= E -> (e-E,e-E).
// ---------------------------------------------------------------------------
template <int H>
__global__ void edge_max(const int* __restrict__ ei, int E, int Nn,
                         const float* __restrict__ es,
                         const float* __restrict__ ed,
                         float* __restrict__ mx) {
    int e = blockIdx.x * blockDim.x + threadIdx.x;
    int ET = E + Nn;
    if (e >= ET) return;
    int s, d;
    if (e < E) { s = ei[e]; d = ei[E + e]; } else { s = d = e - E; }
#pragma unroll
    for (int h = 0; h < H; ++h) {
        float v = es[s * H + h] + ed[d * H + h];
        v = v > 0.f ? v : v * SLOPE;
        __hip_atomic_fetch_max(&mx[d * H + h], v, __ATOMIC_RELAXED,
                               __HIP_MEMORY_SCOPE_AGENT);
    }
}

// ---------------------------------------------------------------------------
// Edge pass B: ex = exp(e - m[dst]); den[dst] += ex; out[dst] += ex * h[src].
// Normalization deferred to finalize (identical to alpha-weighted sum).
// Thread per (edge, head).
// ---------------------------------------------------------------------------
template <int H, int C>
__global__ void edge_aggr(const int* __restrict__ ei, int E, int Nn,
                          const float* __restrict__ es,
                          const float* __restrict__ ed,
                          const float* __restrict__ mx,
                          const float* __restrict__ Hm,
                          float* __restrict__ den, float* __restrict__ out) {
    int t = blockIdx.x * blockDim.x + threadIdx.x;
    int ET = E + Nn;
    if (t >= ET * H) return;
    int e = t / H, h = t % H;
    int s, d;
    if (e < E) { s = ei[e]; d = ei[E + e]; } else { s = d = e - E; }
    float v = es[s * H + h] + ed[d * H + h];
    v = v > 0.f ? v : v * SLOPE;
    float ex = __expf(v - mx[d * H + h]);
    unsafeAtomicAdd(&den[d * H + h], ex);
    const float* hp = Hm + (size_t)s * (H * C) + h * C;
    float* op = out + (size_t)d * (H * C) + h * C;
#pragma unroll
    for (int c = 0; c < C; ++c) unsafeAtomicAdd(&op[c], ex * hp[c]);
}

// Layer-1 finalize: normalize, add bias, ELU (in place -> GEMM2 input).
__global__ void finalize1(float* __restrict__ out, const float* __restrict__ den,
                          const float* __restrict__ b, int Nn) {
    int t = blockIdx.x * blockDim.x + threadIdx.x;
    if (t >= Nn * HID1) return;
    int n = t / HID1, col = t % HID1, h = col / C1C;
    float v = out[t] / den[n * H1H + h] + b[col];
    out[t] = v > 0.f ? v : (__expf(v) - 1.f);
}

// Layer-2 finalize: normalize, add bias, and fold the global sum-pool.
__global__ void finalize2(const float* __restrict__ out, const float* __restrict__ den,
                          const float* __restrict__ b, float* __restrict__ pooled,
                          int Nn) {
    int t = blockIdx.x * blockDim.x + threadIdx.x;
    if (t >= Nn * OUTC) return;
    int n = t / OUTC, c = t % OUTC;
    float v = out[t] / den[n] + b[c];
    unsafeAtomicAdd(&pooled[c], v);
}

// Final scalar: pooled[32] . Wr[32] + br, one wave32 with shuffle reduction.
__global__ __launch_bounds__(32) void final_dot(const float* __restrict__ pooled,
                                                const float* __restrict__ Wr,
                                                const float* __restrict__ br,
                                                float* __restrict__ outp) {
    int lane = threadIdx.x;
    float v = pooled[lane] * Wr[lane];
#pragma unroll
    for (int off = 16; off > 0; off >>= 1) v += __shfl_down(v, off, 32);
    if (lane == 0) outp[0] = v + br[0];
}

// ---------------------------------------------------------------------------
extern "C" void kernel_launch(void* const* d_in, const int* in_sizes, int n_in,
                              void* d_out, int out_size, void* d_ws, size_t ws_size,
                              hipStream_t stream) {
    const float* x     = (const float*)d_in[0];
    const int*   ei    = (const int*)d_in[1];
    const float* W1    = (const float*)d_in[2];
    const float* asrc1 = (const float*)d_in[3];
    const float* adst1 = (const float*)d_in[4];
    const float* b1    = (const float*)d_in[5];
    const float* W2    = (const float*)d_in[6];
    const float* asrc2 = (const float*)d_in[7];
    const float* adst2 = (const float*)d_in[8];
    const float* b2    = (const float*)d_in[9];
    const float* Wr    = (const float*)d_in[10];
    const float* br    = (const float*)d_in[11];

    const int Nn = in_sizes[0] / NFEAT;     // 100000
    const int E  = in_sizes[1] / 2;         // 3200000
    const int ET = E + Nn;                  // + self loops

    // Workspace carve-up (floats).
    float* p = (float*)d_ws;
    float* h1   = p; p += (size_t)Nn * HID1;
    float* out1 = p; p += (size_t)Nn * HID1;
    float* es1  = p; p += (size_t)Nn * H1H;
    float* ed1  = p; p += (size_t)Nn * H1H;
    float* m1   = p; p += (size_t)Nn * H1H;
    float* dn1  = p; p += (size_t)Nn * H1H;
    float* h2   = p; p += (size_t)Nn * OUTC;
    float* out2 = p; p += (size_t)Nn * OUTC;
    float* es2  = p; p += (size_t)Nn;
    float* ed2  = p; p += (size_t)Nn;
    float* m2   = p; p += (size_t)Nn;
    float* dn2  = p; p += (size_t)Nn;
    float* pooled = p; p += OUTC;
    v2f* Wp1 = (v2f*)p; p += (size_t)(NFEAT / 2) * HID1 * 2;
    v2f* Wp2 = (v2f*)p; p += (size_t)(HID1 / 2) * OUTC * 2;

    const int TB = 256;
    const int tiles = (Nn + 15) / 16;

    // Pre-pack weights into K-pair-interleaved float2 (tiny, one shot each).
    pack_w<NFEAT, HID1><<<((NFEAT / 2) * HID1 + TB - 1) / TB, TB, 0, stream>>>(W1, Wp1);
    pack_w<HID1, OUTC><<<((HID1 / 2) * OUTC + TB - 1) / TB, TB, 0, stream>>>(W2, Wp2);

    // ---- Layer 1 ----
    gemm16_wmma<NFEAT, HID1><<<tiles, 32, 0, stream>>>(x, Wp1, h1, Nn);
    escore_init1<<<(Nn * H1H + TB - 1) / TB, TB, 0, stream>>>(
        h1, asrc1, adst1, es1, ed1, m1, dn1, out1, Nn);
    edge_max<H1H><<<(ET + TB - 1) / TB, TB, 0, stream>>>(ei, E, Nn, es1, ed1, m1);
    edge_aggr<H1H, C1C><<<((size_t)ET * H1H + TB - 1) / TB, TB, 0, stream>>>(
        ei, E, Nn, es1, ed1, m1, h1, dn1, out1);
    finalize1<<<(Nn * HID1 + TB - 1) / TB, TB, 0, stream>>>(out1, dn1, b1, Nn);

    // ---- Layer 2 ----
    gemm16_wmma<HID1, OUTC><<<tiles, 32, 0, stream>>>(out1, Wp2, h2, Nn);
    escore_init2<<<(Nn + TB - 1) / TB, TB, 0, stream>>>(
        h2, asrc2, adst2, es2, ed2, m2, dn2, out2, pooled, Nn);
    edge_max<1><<<(ET + TB - 1) / TB, TB, 0, stream>>>(ei, E, Nn, es2, ed2, m2);
    edge_aggr<1, OUTC><<<(ET + TB - 1) / TB, TB, 0, stream>>>(
        ei, E, Nn, es2, ed2, m2, h2, dn2, out2);
    finalize2<<<(Nn * OUTC + TB - 1) / TB, TB, 0, stream>>>(out2, dn2, b2, pooled, Nn);

    // ---- Readout ----
    final_dot<<<1, 32, 0, stream>>>(pooled, Wr, br, (float*)d_out);
}